// MFA_74981539053792
// MI455X (gfx1250) — compile-verified
//
#include <hip/hip_runtime.h>
#include <hip/hip_bf16.h>

// MFA component log-likelihood via Woodbury, fp32 WMMA (V_WMMA_F32_16X16X4_F32).
// n=2048, K=32, d=1024, l=64 (d and l structurally hardcoded; n,K from in_sizes).

typedef __attribute__((ext_vector_type(2))) float v2f;
typedef __attribute__((ext_vector_type(8))) float v8f;

#define LOG2PI_F 1.8378770664093453f

__device__ __forceinline__ v8f wmma4(v2f a, v2f b, v8f c) {
  return __builtin_amdgcn_wmma_f32_16x16x4_f32(false, a, false, b, (short)0, c,
                                               false, false);
}

// ---------------- prep: iD = exp(-log_D), sumlogD[k] ----------------
__global__ __launch_bounds__(256) void k_prep(const float* __restrict__ log_D,
                                              float* __restrict__ iD,
                                              float* __restrict__ sumlogD) {
  __shared__ float red[256];
  int k = blockIdx.x, t = threadIdx.x;
  float s = 0.0f;
  for (int i = t; i < 1024; i += 256) {
    float ld = log_D[k * 1024 + i];
    iD[k * 1024 + i] = __expf(-ld);
    s += ld;
  }
  red[t] = s;
  __syncthreads();
  for (int o = 128; o > 0; o >>= 1) {
    if (t < o) red[t] += red[t + o];
    __syncthreads();
  }
  if (t == 0) sumlogD[k] = red[0];
}

// ---------------- softmax over K (one wave) ----------------
__global__ void k_pi(const float* __restrict__ logits, float* __restrict__ PI,
                     int K) {
  int t = threadIdx.x;  // K==32 threads
  float v = (t < K) ? logits[t] : -3.4e38f;
  float mx = v;
  for (int o = 16; o > 0; o >>= 1) mx = fmaxf(mx, __shfl_xor(mx, o));
  float e = (t < K) ? __expf(v - mx) : 0.0f;
  float ss = e;
  for (int o = 16; o > 0; o >>= 1) ss += __shfl_xor(ss, o);
  if (t < K) PI[t] = e / ss;
}

// ---------------- L[k] = I + (iD*A)^T A  (64x64, inner d=1024) ----------------
#define GS 76  // LDS row stride (floats): conflict-free for both operand patterns
__global__ __launch_bounds__(256) void k_gram(const float* __restrict__ A,
                                              const float* __restrict__ iD,
                                              float* __restrict__ Lbuf) {
  __shared__ float sA[64 * GS];   // raw A chunk   [dd][lc]
  __shared__ float sAs[64 * GS];  // iD-scaled A   [dd][lc]
  int k = blockIdx.x, t = threadIdx.x;
  int lane = t & 31, wv = t >> 5;
  int g = lane >> 4, m = lane & 15;
  int rr = t >> 2, cb = (t & 3) * 16;
  int t0 = wv * 2, t1 = wv * 2 + 1;
  int ti0 = t0 >> 2, tj0 = t0 & 3, ti1 = t1 >> 2, tj1 = t1 & 3;
  v8f acc0 = {}, acc1 = {};

  for (int d0 = 0; d0 < 1024; d0 += 64) {
    const float* Ar = A + ((size_t)k * 1024 + d0 + rr) * 64 + cb;
    float sc = iD[k * 1024 + d0 + rr];
#pragma unroll
    for (int j = 0; j < 16; ++j) {
      float av = Ar[j];
      sA[rr * GS + cb + j] = av;
      sAs[rr * GS + cb + j] = av * sc;
    }
    __syncthreads();
#pragma unroll
    for (int kk = 0; kk < 64; kk += 4) {
      int kr = kk + 2 * g;
      v2f a0, b0, a1, b1;
      a0.x = sAs[(kr + 0) * GS + ti0 * 16 + m];  // ((iD*A)^T)[M][K] = sAs[K][M]
      a0.y = sAs[(kr + 1) * GS + ti0 * 16 + m];
      b0.x = sA[(kr + 0) * GS + tj0 * 16 + m];   // A[K][N]
      b0.y = sA[(kr + 1) * GS + tj0 * 16 + m];
      acc0 = wmma4(a0, b0, acc0);
      a1.x = sAs[(kr + 0) * GS + ti1 * 16 + m];
      a1.y = sAs[(kr + 1) * GS + ti1 * 16 + m];
      b1.x = sA[(kr + 0) * GS + tj1 * 16 + m];
      b1.y = sA[(kr + 1) * GS + tj1 * 16 + m];
      acc1 = wmma4(a1, b1, acc1);
    }
    __syncthreads();
  }
  float* Lk = Lbuf + (size_t)k * 4096;
#pragma unroll
  for (int v = 0; v < 8; ++v) {
    int r0 = ti0 * 16 + v + 8 * g, c0 = tj0 * 16 + m;
    Lk[r0 * 64 + c0] = acc0[v] + ((r0 == c0) ? 1.0f : 0.0f);
    int r1 = ti1 * 16 + v + 8 * g, c1 = tj1 * 16 + m;
    Lk[r1 * 64 + c1] = acc1[v] + ((r1 == c1) ? 1.0f : 0.0f);
  }
}

// -------- iL[k] = inv(L[k]) (Gauss-Jordan, SPD -> no pivoting), logdet --------
__global__ __launch_bounds__(256) void k_inv(const float* __restrict__ Lbuf,
                                             const float* __restrict__ sumlogD,
                                             const float* __restrict__ PI,
                                             float* __restrict__ iLbuf,
                                             float* __restrict__ cterm) {
  __shared__ float M[64 * 65];
  __shared__ float Inv[64 * 65];
  __shared__ float fcol[64];
  int k = blockIdx.x, t = threadIdx.x;
  const float* Lk = Lbuf + (size_t)k * 4096;
  for (int i = t; i < 4096; i += 256) {
    int r = i >> 6, c = i & 63;
    M[r * 65 + c] = Lk[i];
    Inv[r * 65 + c] = (r == c) ? 1.0f : 0.0f;
  }
  __syncthreads();
  float slog = 0.0f;  // only thread 0's copy is used
  for (int p = 0; p < 64; ++p) {
    if (t < 64) fcol[t] = M[t * 65 + p];
    __syncthreads();
    float pv = fcol[p];
    float pinv = 1.0f / pv;
    if (t == 0) slog += logf(pv);
    if (t < 128) {
      if (t < 64) M[p * 65 + t] *= pinv;
      else Inv[p * 65 + (t - 64)] *= pinv;
    }
    __syncthreads();
    for (int i = t; i < 8192; i += 256) {
      int r = i >> 7, c = i & 127;
      if (r != p) {
        float f = fcol[r];
        if (c < 64) M[r * 65 + c] -= f * M[p * 65 + c];
        else Inv[r * 65 + (c - 64)] -= f * Inv[p * 65 + (c - 64)];
      }
    }
    __syncthreads();
  }
  float* iLk = iLbuf + (size_t)k * 4096;
  for (int i = t; i < 4096; i += 256) iLk[i] = Inv[(i >> 6) * 65 + (i & 63)];
  if (t == 0) {
    float logdet = slog + sumlogD[k];
    cterm[k] = PI[k] - 0.5f * (1024.0f * LOG2PI_F + logdet);
  }
}

// -------- main: per (k, 64-row tile): u=(xc*iD)@A, m_d, output --------
__global__ __launch_bounds__(256) void k_main(const float* __restrict__ x,
                                              const float* __restrict__ MU,
                                              const float* __restrict__ A,
                                              const float* __restrict__ iD,
                                              const float* __restrict__ iLbuf,
                                              const float* __restrict__ cterm,
                                              float* __restrict__ out, int K) {
  __shared__ float sXI[64 * GS];  // phase A: (x-mu)*iD chunk; phase B: u matrix
  __shared__ float sB[64 * GS];   // phase A: A chunk;         phase B: iL
  __shared__ float sq1[64];
  __shared__ float sm2[64];
  int k = blockIdx.y;
  int row0 = blockIdx.x * 64;
  int t = threadIdx.x, lane = t & 31, wv = t >> 5;
  int g = lane >> 4, m = lane & 15;
  int t0 = wv * 2, t1 = wv * 2 + 1;
  int ti0 = t0 >> 2, tj0 = t0 & 3, ti1 = t1 >> 2, tj1 = t1 & 3;
  int rr = t >> 2, cb = (t & 3) * 16;
  v8f u0 = {}, u1 = {};
  float q1p = 0.0f;
  if (t < 64) sm2[t] = 0.0f;

  for (int d0 = 0; d0 < 1024; d0 += 64) {
    const float* xr = x + (size_t)(row0 + rr) * 1024 + d0 + cb;
    const float* mr = MU + (size_t)k * 1024 + d0 + cb;
    const float* dr = iD + (size_t)k * 1024 + d0 + cb;
    const float* Ar = A + ((size_t)k * 1024 + d0 + rr) * 64 + cb;
#pragma unroll
    for (int j = 0; j < 16; ++j) {
      float xc = xr[j] - mr[j];
      float xi = xc * dr[j];
      q1p += xc * xi;  // xc^2 * iD
      sXI[rr * GS + cb + j] = xi;
      sB[rr * GS + cb + j] = Ar[j];
    }
    __syncthreads();
#pragma unroll
    for (int kk = 0; kk < 64; kk += 4) {
      int kr = kk + 2 * g;
      v2f a0, b0, a1, b1;
      a0.x = sXI[(ti0 * 16 + m) * GS + kr];
      a0.y = sXI[(ti0 * 16 + m) * GS + kr + 1];
      b0.x = sB[(kr + 0) * GS + tj0 * 16 + m];
      b0.y = sB[(kr + 1) * GS + tj0 * 16 + m];
      u0 = wmma4(a0, b0, u0);
      a1.x = sXI[(ti1 * 16 + m) * GS + kr];
      a1.y = sXI[(ti1 * 16 + m) * GS + kr + 1];
      b1.x = sB[(kr + 0) * GS + tj1 * 16 + m];
      b1.y = sB[(kr + 1) * GS + tj1 * 16 + m];
      u1 = wmma4(a1, b1, u1);
    }
    __syncthreads();
  }

  // q1 per row: reduce across the 4 threads of each row-quad
  q1p += __shfl_xor(q1p, 1);
  q1p += __shfl_xor(q1p, 2);
  if ((t & 3) == 0) sq1[rr] = q1p;

  // spill u tiles into sXI as Us[row][col]; load iL into sB
#pragma unroll
  for (int v = 0; v < 8; ++v) {
    sXI[(ti0 * 16 + v + 8 * g) * GS + tj0 * 16 + m] = u0[v];
    sXI[(ti1 * 16 + v + 8 * g) * GS + tj1 * 16 + m] = u1[v];
  }
  const float* iLk = iLbuf + (size_t)k * 4096;
  for (int i = t; i < 4096; i += 256) sB[(i >> 6) * GS + (i & 63)] = iLk[i];
  __syncthreads();

  // T = Us @ iL  (WMMA), then m2[row] = sum_col T[row][col]*Us[row][col]
  v8f c0 = {}, c1 = {};
#pragma unroll
  for (int kk = 0; kk < 64; kk += 4) {
    int kr = kk + 2 * g;
    v2f a0, b0, a1, b1;
    a0.x = sXI[(ti0 * 16 + m) * GS + kr];
    a0.y = sXI[(ti0 * 16 + m) * GS + kr + 1];
    b0.x = sB[(kr + 0) * GS + tj0 * 16 + m];
    b0.y = sB[(kr + 1) * GS + tj0 * 16 + m];
    c0 = wmma4(a0, b0, c0);
    a1.x = sXI[(ti1 * 16 + m) * GS + kr];
    a1.y = sXI[(ti1 * 16 + m) * GS + kr + 1];
    b1.x = sB[(kr + 0) * GS + tj1 * 16 + m];
    b1.y = sB[(kr + 1) * GS + tj1 * 16 + m];
    c1 = wmma4(a1, b1, c1);
  }
#pragma unroll
  for (int v = 0; v < 8; ++v) {
    int r0 = ti0 * 16 + v + 8 * g, cc0 = tj0 * 16 + m;
    float pv = c0[v] * sXI[r0 * GS + cc0];
    pv += __shfl_xor(pv, 1); pv += __shfl_xor(pv, 2);
    pv += __shfl_xor(pv, 4); pv += __shfl_xor(pv, 8);
    if (m == 0) atomicAdd(&sm2[r0], pv);
    int r1 = ti1 * 16 + v + 8 * g, cc1 = tj1 * 16 + m;
    float qv = c1[v] * sXI[r1 * GS + cc1];
    qv += __shfl_xor(qv, 1); qv += __shfl_xor(qv, 2);
    qv += __shfl_xor(qv, 4); qv += __shfl_xor(qv, 8);
    if (m == 0) atomicAdd(&sm2[r1], qv);
  }
  __syncthreads();
  if (t < 64) {
    float md = sq1[t] - sm2[t];
    out[(size_t)(row0 + t) * K + k] = cterm[k] - 0.5f * md;
  }
}

extern "C" void kernel_launch(void* const* d_in, const int* in_sizes, int n_in,
                              void* d_out, int out_size, void* d_ws,
                              size_t ws_size, hipStream_t stream) {
  const float* x   = (const float*)d_in[0];   // [n, 1024]
  const float* MU  = (const float*)d_in[1];   // [K, 1024]
  const float* A   = (const float*)d_in[2];   // [K, 1024, 64]
  const float* lgD = (const float*)d_in[3];   // [K, 1024]
  const float* PIl = (const float*)d_in[4];   // [K]
  float* out = (float*)d_out;

  int n = in_sizes[0] / 1024;
  int K = in_sizes[4];

  float* ws = (float*)d_ws;
  float* w_iD   = ws;                          // K*1024
  float* w_L    = w_iD + (size_t)K * 1024;     // K*4096
  float* w_iL   = w_L + (size_t)K * 4096;      // K*4096
  float* w_slD  = w_iL + (size_t)K * 4096;     // K
  float* w_PI   = w_slD + K;                   // K
  float* w_ct   = w_PI + K;                    // K
  (void)ws_size; (void)n_in; (void)out_size;

  k_pi<<<1, 32, 0, stream>>>(PIl, w_PI, K);
  k_prep<<<K, 256, 0, stream>>>(lgD, w_iD, w_slD);
  k_gram<<<K, 256, 0, stream>>>(A, w_iD, w_L);
  k_inv<<<K, 256, 0, stream>>>(w_L, w_slD, w_PI, w_iL, w_ct);
  dim3 grid(n / 64, K);
  k_main<<<grid, 256, 0, stream>>>(x, MU, A, w_iD, w_iL, w_ct, out, K);
}